// MiT_55379308315308
// MI455X (gfx1250) — compile-verified
//
#include <hip/hip_runtime.h>
#include <hip/hip_bf16.h>

// ---------------------------------------------------------------------------
// CDNA5 fp32 WMMA types (16x16x4 f32: A,B = 2 VGPRs; C/D = 8 VGPRs per lane)
// ---------------------------------------------------------------------------
typedef __attribute__((ext_vector_type(2))) float v2f;
typedef __attribute__((ext_vector_type(8))) float v8f;
typedef __attribute__((ext_vector_type(4))) unsigned int v4u;
typedef __attribute__((ext_vector_type(8))) int v8i;
typedef __attribute__((ext_vector_type(4))) int v4i;

__device__ __forceinline__ v8f wmma_f32_4(v2f a, v2f b, v8f c) {
    return __builtin_amdgcn_wmma_f32_16x16x4_f32(false, a, false, b, (short)0, c,
                                                 false, false);
}

#define FLAG_BIAS 1
#define FLAG_GELU 2
#define FLAG_RES  4

// ---------------------------------------------------------------------------
// 1x1-conv GEMM with TDM-staged B operand.
//   Out[b,co,hw] = sum_ci W[co,ci] * X[b,ci,hw]   (+bias/gelu/residual)
// Block = 4 waves. Block owns one 16-column N tile; wave w owns M tile
// (blockIdx.x*4 + w). The X panel chunk (Kc=32 rows x 16 cols, row stride HW)
// is DMA'd into LDS once per chunk by the Tensor Data Mover and shared by all
// 4 waves; WMMA reads B from LDS, A (weight rows) from global.
// Every Cin in this net is a multiple of 32; B*HW is a multiple of 16.
// ---------------------------------------------------------------------------
__global__ __launch_bounds__(128)
void gemm1x1_f32_wmma(const float* __restrict__ W, const float* __restrict__ X,
                      const float* __restrict__ bias, const float* __restrict__ res,
                      float* __restrict__ Out,
                      int Cout, int Cin, int HW, int B, int flags)
{
    __shared__ float Bt[32 * 16];                 // one K-chunk of the X panel

    const int lane   = threadIdx.x;               // 0..31
    const int wy     = threadIdx.y;               // 0..3
    const int mTile  = blockIdx.x * 4 + wy;
    const int Mtiles = Cout >> 4;
    const bool active = (mTile < Mtiles);         // wave-uniform

    const int m    = lane & 15;
    const int hi   = lane >> 4;                   // 0 or 1
    const int koff = hi * 2;

    const int nTile = blockIdx.y;
    const int nG    = (nTile << 4) + m;           // this lane's global column
    const int bN    = nG / HW;
    const int hw    = nG - bN * HW;

    // block-uniform tile origin for the TDM descriptor
    const int nG0 = nTile << 4;
    const int b0  = nG0 / HW;
    const int hw0 = nG0 - b0 * HW;

    // D# group 1: data_size=4B(code 2); tensor 16 x 32; tile 16 x 32;
    // tensor_dim0_stride = HW elements (row-to-row channel stride of X).
    v8i g1;
    g1[0] = 0x20000;            // workgroup_mask=0, data_size=2 (4 bytes)
    g1[1] = (16 << 16);         // tensor_dim0 = 16    (bits 79:48)
    g1[2] = (32 << 16);         // tensor_dim1 = 32    (bits 111:80)
    g1[3] = (16 << 16);         // tile_dim0   = 16    (bits 127:112)
    g1[4] = 32;                 // tile_dim1   = 32, tile_dim2 = 0
    g1[5] = HW;                 // tensor_dim0_stride[31:0]
    g1[6] = 0;
    g1[7] = 0;
    const v4i gz4 = {0, 0, 0, 0};
    const v8i gz8 = {0, 0, 0, 0, 0, 0, 0, 0};
    const unsigned ldsOff = (unsigned)(uintptr_t)(&Bt[0]);   // LDS byte offset

    const float* Wrow = active ? (W + (size_t)(mTile * 16 + m) * Cin) : W;

    v8f acc = {};
    for (int k0 = 0; k0 < Cin; k0 += 32) {
        __syncthreads();                          // chunk consumed by all waves
        if (wy == 0) {
            const unsigned long long ga = (unsigned long long)(uintptr_t)
                (X + ((size_t)b0 * Cin + k0) * HW + hw0);
            v4u g0;
            g0[0] = 1u;                           // count=1, user descriptor
            g0[1] = ldsOff;                       // lds_addr
            g0[2] = (unsigned)ga;                 // global_addr[31:0]
            g0[3] = (unsigned)((ga >> 32) & 0x01FFFFFFu) | (2u << 30); // type=2
            __builtin_amdgcn_tensor_load_to_lds(g0, g1, gz4, gz4, gz8, 0);
            __builtin_amdgcn_s_wait_tensorcnt(0); // TDM done -> LDS valid
        }
        __syncthreads();                          // publish chunk to all waves

        if (active) {
            #pragma unroll
            for (int kk = 0; kk < 32; kk += 4) {
                v2f a, b;
                a.x = Wrow[k0 + kk + koff];
                a.y = Wrow[k0 + kk + koff + 1];
                b.x = Bt[(kk + koff) * 16 + m];
                b.y = Bt[(kk + koff + 1) * 16 + m];
                acc = wmma_f32_4(a, b, acc);
            }
        }
    }

    if (!active) return;
    const int baseRow = mTile * 16 + hi * 8;
    #pragma unroll
    for (int r = 0; r < 8; ++r) {
        int row = baseRow + r;
        float v = acc[r];
        if (flags & FLAG_BIAS) v += bias[row];
        if (flags & FLAG_GELU) v = 0.5f * v * (1.0f + erff(v * 0.70710678118654752f));
        size_t o = (size_t)bN * Cout * HW + (size_t)row * HW + hw;
        if (flags & FLAG_RES) v += res[o];
        Out[o] = v;
    }
}

// ---------------------------------------------------------------------------
// Image-branch attention.  For every stage: head_dim d = 32, KV length = 64.
// One wave handles one (batch*head, 16-row Q tile):
//   S(16x64) = Q(16x32) @ K^T(32x64)  -> 4 N-tiles x 8 WMMA steps
//   softmax rows (LDS)
//   O(16x32) = P(16x64) @ V(64x32)    -> 2 N-tiles x 16 WMMA steps
// ---------------------------------------------------------------------------
__global__ __launch_bounds__(128)
void attn_img_wmma(const float* __restrict__ Q, const float* __restrict__ KV,
                   float* __restrict__ Out, int dim, int heads, int HW, float scale)
{
    __shared__ float Sm[4][16][68];
    const int lane  = threadIdx.x;
    const int w     = threadIdx.y;
    const int qTile = blockIdx.y * 4 + w;                // grid sized exactly
    const int bh    = blockIdx.x;
    const int b     = bh / heads;
    const int h     = bh - b * heads;

    const int m    = lane & 15;
    const int hi   = lane >> 4;
    const int koff = hi * 2;
    const int hwq  = qTile * 16 + m;

    const float* Qb = Q  + ((size_t)b * dim + h * 32) * HW + hwq;       // + c*HW
    const float* Kb = KV + ((size_t)b * 2 * dim + h * 32) * 64;         // + c*64 + nk
    const float* Vb = KV + ((size_t)b * 2 * dim + dim + h * 32) * 64;   // + c*64 + nk

    // ---- S = Q @ K^T --------------------------------------------------------
    v8f s0 = {}, s1 = {}, s2 = {}, s3 = {};
    for (int c = 0; c < 32; c += 4) {
        v2f a;
        a.x = Qb[(size_t)(c + koff) * HW];
        a.y = Qb[(size_t)(c + koff + 1) * HW];
        const float* K0 = Kb + (size_t)(c + koff) * 64;
        const float* K1 = K0 + 64;
        v2f b0; b0.x = K0[m];      b0.y = K1[m];
        v2f b1; b1.x = K0[16 + m]; b1.y = K1[16 + m];
        v2f b2; b2.x = K0[32 + m]; b2.y = K1[32 + m];
        v2f b3; b3.x = K0[48 + m]; b3.y = K1[48 + m];
        s0 = wmma_f32_4(a, b0, s0);
        s1 = wmma_f32_4(a, b1, s1);
        s2 = wmma_f32_4(a, b2, s2);
        s3 = wmma_f32_4(a, b3, s3);
    }

    float (*S)[68] = Sm[w];
    #pragma unroll
    for (int r = 0; r < 8; ++r) {
        int row = hi * 8 + r;
        S[row][ 0 + m] = s0[r] * scale;
        S[row][16 + m] = s1[r] * scale;
        S[row][32 + m] = s2[r] * scale;
        S[row][48 + m] = s3[r] * scale;
    }
    __syncthreads();

    // ---- row softmax (lanes 0..15 each own one row of 64) -------------------
    if (lane < 16) {
        float mx = -1e30f;
        for (int j = 0; j < 64; ++j) mx = fmaxf(mx, S[lane][j]);
        float sum = 0.f;
        for (int j = 0; j < 64; ++j) { float e = expf(S[lane][j] - mx); S[lane][j] = e; sum += e; }
        float inv = 1.0f / sum;
        for (int j = 0; j < 64; ++j) S[lane][j] *= inv;
    }
    __syncthreads();

    // ---- O = P @ V ----------------------------------------------------------
    v8f o0 = {}, o1 = {};
    for (int k = 0; k < 64; k += 4) {
        v2f a;
        a.x = S[m][k + koff];
        a.y = S[m][k + koff + 1];
        v2f b0; b0.x = Vb[(size_t)( 0 + m) * 64 + k + koff];
                b0.y = Vb[(size_t)( 0 + m) * 64 + k + koff + 1];
        v2f b1; b1.x = Vb[(size_t)(16 + m) * 64 + k + koff];
                b1.y = Vb[(size_t)(16 + m) * 64 + k + koff + 1];
        o0 = wmma_f32_4(a, b0, o0);
        o1 = wmma_f32_4(a, b1, o1);
    }

    float* Ob = Out + ((size_t)b * dim + h * 32) * HW;   // + c*HW + hwq
    #pragma unroll
    for (int r = 0; r < 8; ++r) {
        int qrow = qTile * 16 + hi * 8 + r;
        Ob[(size_t)( 0 + m) * HW + qrow] = o0[r];
        Ob[(size_t)(16 + m) * HW + qrow] = o1[r];
    }
}

// ---------------------------------------------------------------------------
// Batch-branch attention: softmax over the batch dim (len 8) per (hw, head).
// Tiny (8x32 per group) -> scalar; ACCUMULATES into Out (after attn_img).
// ---------------------------------------------------------------------------
__global__ void attn_batch(const float* __restrict__ QB, const float* __restrict__ KVB,
                           float* __restrict__ Out, int dim, int heads, int HW, int B,
                           float scale)
{
    int t = blockIdx.x * blockDim.x + threadIdx.x;
    int total = B * heads * HW;
    if (t >= total) return;
    int hw = t % HW; int tmp = t / HW;
    int h = tmp % heads; int bq = tmp / heads;

    float s[8];
    for (int b2 = 0; b2 < B; ++b2) {
        float acc = 0.f;
        for (int c = 0; c < 32; ++c) {
            float qv = QB [((size_t)bq * dim + h * 32 + c) * HW + hw];
            float kv = KVB[((size_t)b2 * 2 * dim + h * 32 + c) * HW + hw];
            acc += qv * kv;
        }
        s[b2] = acc * scale;
    }
    float mx = s[0];
    for (int i = 1; i < B; ++i) mx = fmaxf(mx, s[i]);
    float sum = 0.f;
    for (int i = 0; i < B; ++i) { s[i] = expf(s[i] - mx); sum += s[i]; }
    float inv = 1.0f / sum;
    for (int i = 0; i < B; ++i) s[i] *= inv;

    for (int c = 0; c < 32; ++c) {
        float acc = 0.f;
        for (int b2 = 0; b2 < B; ++b2)
            acc += s[b2] * KVB[((size_t)b2 * 2 * dim + dim + h * 32 + c) * HW + hw];
        size_t o = ((size_t)bq * dim + h * 32 + c) * HW + hw;
        Out[o] += acc;
    }
}

// ---------------------------------------------------------------------------
// Channel LayerNorm, reference semantics: (x-mean)/(sqrt(var_biased)+eps)*g+b
// ---------------------------------------------------------------------------
__global__ void layernorm_ch(const float* __restrict__ X, const float* __restrict__ g,
                             const float* __restrict__ bta, float* __restrict__ Out,
                             int C, int HW, int B)
{
    int t = blockIdx.x * blockDim.x + threadIdx.x;
    if (t >= B * HW) return;
    int b = t / HW, hw = t - b * HW;
    const float* xb = X + (size_t)b * C * HW + hw;
    float mean = 0.f;
    for (int c = 0; c < C; ++c) mean += xb[(size_t)c * HW];
    mean /= (float)C;
    float var = 0.f;
    for (int c = 0; c < C; ++c) { float d = xb[(size_t)c * HW] - mean; var += d * d; }
    var /= (float)C;
    float inv = 1.0f / (sqrtf(var) + 1e-5f);
    float* ob = Out + (size_t)b * C * HW + hw;
    for (int c = 0; c < C; ++c)
        ob[(size_t)c * HW] = (xb[(size_t)c * HW] - mean) * inv * g[c] + bta[c];
}

// ---------------------------------------------------------------------------
// Generic direct conv (patch embeds, rr x rr KV-reduce). Scalar: tiny FLOPs.
// ---------------------------------------------------------------------------
__global__ void conv2d_direct(const float* __restrict__ X, const float* __restrict__ W,
                              const float* __restrict__ bias, float* __restrict__ Out,
                              int B, int Cin, int Hin, int Win,
                              int Cout, int Hout, int Wout, int K, int S, int P)
{
    int t = blockIdx.x * blockDim.x + threadIdx.x;
    int total = B * Cout * Hout * Wout;
    if (t >= total) return;
    int ow = t % Wout; int tmp = t / Wout;
    int oh = tmp % Hout; tmp /= Hout;
    int co = tmp % Cout; int b = tmp / Cout;

    float acc = bias ? bias[co] : 0.f;
    const float* wb = W + (size_t)co * Cin * K * K;
    for (int ci = 0; ci < Cin; ++ci) {
        const float* xb = X + ((size_t)b * Cin + ci) * Hin * Win;
        for (int kh = 0; kh < K; ++kh) {
            int ih = oh * S - P + kh;
            if (ih < 0 || ih >= Hin) continue;
            for (int kw = 0; kw < K; ++kw) {
                int iw = ow * S - P + kw;
                if (iw < 0 || iw >= Win) continue;
                acc += xb[(size_t)ih * Win + iw] * wb[(ci * K + kh) * K + kw];
            }
        }
    }
    Out[t] = acc;
}

// ---------------------------------------------------------------------------
// Depthwise 3x3, pad 1, + bias
// ---------------------------------------------------------------------------
__global__ void dwconv3x3(const float* __restrict__ X, const float* __restrict__ W,
                          const float* __restrict__ bias, float* __restrict__ Out,
                          int B, int C, int H, int Wd)
{
    int t = blockIdx.x * blockDim.x + threadIdx.x;
    int total = B * C * H * Wd;
    if (t >= total) return;
    int ow = t % Wd; int tmp = t / Wd;
    int oh = tmp % H; tmp /= H;
    int c = tmp % C; int b = tmp / C;

    float acc = bias[c];
    const float* xb = X + ((size_t)b * C + c) * H * Wd;
    const float* wb = W + (size_t)c * 9;
    for (int kh = 0; kh < 3; ++kh) {
        int ih = oh - 1 + kh;
        if (ih < 0 || ih >= H) continue;
        for (int kw = 0; kw < 3; ++kw) {
            int iw = ow - 1 + kw;
            if (iw < 0 || iw >= Wd) continue;
            acc += xb[(size_t)ih * Wd + iw] * wb[kh * 3 + kw];
        }
    }
    Out[t] = acc;
}

// ---------------------------------------------------------------------------
// Host orchestration
// ---------------------------------------------------------------------------
extern "C" void kernel_launch(void* const* d_in, const int* in_sizes, int n_in,
                              void* d_out, int out_size, void* d_ws, size_t ws_size,
                              hipStream_t stream)
{
    (void)in_sizes; (void)n_in; (void)ws_size;
    const int B = 8;
    static const int DIMS[4]  = {32, 64, 160, 256};
    static const int HEADS[4] = {1, 2, 5, 8};
    static const int FFEXP[4] = {8, 8, 4, 4};
    static const int RRv[4]   = {8, 4, 2, 1};
    static const int Kk[4]    = {7, 3, 3, 3};
    static const int Ss[4]    = {4, 2, 2, 2};
    static const int Pp[4]    = {3, 1, 1, 1};
    const float scale = 0.1767766952966369f;   // 32^-0.5

    int idx = 0;
    const float* x_in = (const float*)d_in[idx++];

    // workspace layout (floats)
    const size_t A_MAX   = (size_t)8 * 32 * 64 * 64;   // 1,048,576
    const size_t KV_MAX  = (size_t)8 * 2 * 256 * 64;   //   262,144
    const size_t KVB_MAX = 2 * A_MAX;                  // 2,097,152
    const size_t H_MAX   = (size_t)8 * 256 * 64 * 64;  // 8,388,608
    float* ws  = (float*)d_ws;
    float* X0  = ws;
    float* X1  = X0  + A_MAX;
    float* LN  = X1  + A_MAX;
    float* Qb  = LN  + A_MAX;
    float* QBb = Qb  + A_MAX;
    float* AT  = QBb + A_MAX;
    float* KV  = AT  + A_MAX;
    float* KVB = KV  + KV_MAX;
    float* H1  = KVB + KVB_MAX;
    float* DW  = H1  + H_MAX;

    float* cur = X0;
    float* oth = X1;

    int Cin = 3, Hin = 256, Win = 256;

    for (int st = 0; st < 4; ++st) {
        const float* embed_w = (const float*)d_in[idx++];
        const float* embed_b = (const float*)d_in[idx++];
        const float* Lp[2][17];
        for (int l = 0; l < 2; ++l)
            for (int j = 0; j < 17; ++j)
                Lp[l][j] = (const float*)d_in[idx++];

        const int dim   = DIMS[st];
        const int heads = HEADS[st];
        const int rr    = RRv[st];
        const int hid   = dim * FFEXP[st];
        const int Hout  = (Hin + 2 * Pp[st] - Kk[st]) / Ss[st] + 1;
        const int Wout  = (Win + 2 * Pp[st] - Kk[st]) / Ss[st] + 1;
        const int HW    = Hout * Wout;

        // ---- patch embed conv ----
        {
            const float* src = (st == 0) ? x_in : cur;
            int total = B * dim * HW;
            conv2d_direct<<<(total + 255) / 256, 256, 0, stream>>>(
                src, embed_w, embed_b, oth,
                B, Cin, Hin, Win, dim, Hout, Wout, Kk[st], Ss[st], Pp[st]);
            float* t = cur; cur = oth; oth = t;
        }

        const int ntiles = (B * HW) >> 4;
        const dim3 gblk(32, 4);
        auto ggrid = [&](int Cout) { return dim3((Cout / 16 + 3) / 4, ntiles); };

        for (int l = 0; l < 2; ++l) {
            const float* const* L = Lp[l];
            // ======== attention block ========
            layernorm_ch<<<(B * HW + 255) / 256, 256, 0, stream>>>(
                cur, L[0], L[1], LN, dim, HW, B);

            // q = wq(ln)
            gemm1x1_f32_wmma<<<ggrid(dim), gblk, 0, stream>>>(
                L[2], LN, nullptr, nullptr, Qb, dim, dim, HW, B, 0);
            // kv = wkv(ln), stride rr, kernel rr  ->  (B, 2dim, 8, 8)
            {
                int total = B * 2 * dim * 64;
                conv2d_direct<<<(total + 255) / 256, 256, 0, stream>>>(
                    LN, L[3], nullptr, KV,
                    B, dim, Hout, Wout, 2 * dim, Hout / rr, Wout / rr, rr, rr, 0);
            }
            // batch-branch projections (1x1)
            gemm1x1_f32_wmma<<<ggrid(dim), gblk, 0, stream>>>(
                L[5], LN, nullptr, nullptr, QBb, dim, dim, HW, B, 0);
            gemm1x1_f32_wmma<<<ggrid(2 * dim), gblk, 0, stream>>>(
                L[6], LN, nullptr, nullptr, KVB, 2 * dim, dim, HW, B, 0);

            // image attention (writes AT), batch attention (adds into AT)
            attn_img_wmma<<<dim3(B * heads, (HW / 16) / 4), gblk, 0, stream>>>(
                Qb, KV, AT, dim, heads, HW, scale);
            {
                int total = B * heads * HW;
                attn_batch<<<(total + 255) / 256, 256, 0, stream>>>(
                    QBb, KVB, AT, dim, heads, HW, B, scale);
            }

            // out = wo(AT) + x   (residual)
            gemm1x1_f32_wmma<<<ggrid(dim), gblk, 0, stream>>>(
                L[4], AT, nullptr, cur, oth, dim, dim, HW, B, FLAG_RES);
            { float* t = cur; cur = oth; oth = t; }

            // ======== MixFFN block ========
            layernorm_ch<<<(B * HW + 255) / 256, 256, 0, stream>>>(
                cur, L[7], L[8], LN, dim, HW, B);
            // h1 = w1(ln) + b1
            gemm1x1_f32_wmma<<<ggrid(hid), gblk, 0, stream>>>(
                L[9], LN, L[10], nullptr, H1, hid, dim, HW, B, FLAG_BIAS);
            // dw = dwconv3x3(h1) + db
            {
                int total = B * hid * HW;
                dwconv3x3<<<(total + 255) / 256, 256, 0, stream>>>(
                    H1, L[11], L[12], DW, B, hid, Hout, Wout);
            }
            // g = gelu(pw(dw) + pb)
            gemm1x1_f32_wmma<<<ggrid(hid), gblk, 0, stream>>>(
                L[13], DW, L[14], nullptr, H1, hid, hid, HW, B,
                FLAG_BIAS | FLAG_GELU);
            // out = w2(g) + b2 + x
            gemm1x1_f32_wmma<<<ggrid(dim), gblk, 0, stream>>>(
                L[15], H1, L[16], cur, oth, dim, hid, HW, B,
                FLAG_BIAS | FLAG_RES);
            { float* t = cur; cur = oth; oth = t; }
        }

        Cin = dim; Hin = Hout; Win = Wout;
    }

    (void)hipMemcpyAsync(d_out, cur, (size_t)out_size * sizeof(float),
                         hipMemcpyDeviceToDevice, stream);
}